// PNCC_1924145348854
// MI455X (gfx1250) — compile-verified
//
#include <hip/hip_runtime.h>
#include <math.h>

// ---------------- problem constants ----------------
#define BATCH   32
#define LSIG    160000
#define NFFT    1024
#define HOP     160
#define NFRAMES 1001                 // (160000 + 1024 - 1024)/160 + 1
#define FBINS   513                  // NFFT/2 + 1
#define FPAD    520                  // FBINS padded to multiple of 4 (K for WMMA)
#define NFILT   40
#define NPAD    48                   // N padded to 3 x 16 tiles
#define MROWS   (BATCH * NFRAMES)    // 32032 = 2002 * 16
#define MTILES  (MROWS / 16)         // 2002
#define KMAX    FPAD                 // max K staged in LDS

#define LM_A 0.999f
#define LM_B 0.5f
#define LM_T 0.85f
#define LM_MU 0.999f
#define MU_T 0.2f
#define CC   2.0f

typedef __attribute__((ext_vector_type(2))) float        v2f;
typedef __attribute__((ext_vector_type(8))) float        v8f;
typedef __attribute__((ext_vector_type(4))) unsigned int u32x4;
typedef __attribute__((ext_vector_type(8))) int          i32x8;
typedef __attribute__((ext_vector_type(4))) int          i32x4;

// ---------------------------------------------------------------------------
// Kernel 1: fused preemphasis + framing + Hamming window + 1024-pt Stockham
// FFT (complex, real input) + power spectrum. One 128-thread block per frame.
// Output row padded to FPAD floats (tail zeroed) for the WMMA GEMM.
// ---------------------------------------------------------------------------
__global__ void __launch_bounds__(128)
stft_power_kernel(const float* __restrict__ x, const float* __restrict__ win,
                  float* __restrict__ spec)
{
    __shared__ float2 bufA[NFFT];
    __shared__ float2 bufB[NFFT];

    const int fr  = blockIdx.x;
    const int b   = fr / NFRAMES;
    const int t   = fr - b * NFRAMES;
    const float* xb = x + (size_t)b * LSIG;
    const int tid = threadIdx.x;

    // load frame: xe[s] = (s==0 ? x[0] : x[s] - 0.97*x[s-1]); zero outside [0,L)
    for (int i = tid; i < NFFT; i += 128) {
        int s = t * HOP + i - NFFT / 2;     // center padding of NFFT/2
        float v = 0.0f;
        if (s >= 0 && s < LSIG) {
            float cur = xb[s];
            v = (s == 0) ? cur : fmaf(-0.97f, xb[s - 1], cur);
        }
        bufA[i] = make_float2(v * win[i], 0.0f);
    }
    __syncthreads();

    // Stockham radix-2 DIF (auto-sorted), 10 stages, ping-pong in LDS
    float2* X = bufA;
    float2* Y = bufB;
    int n = NFFT, s = 1;
    #pragma unroll 1
    for (int stage = 0; stage < 10; ++stage) {
        const int   m   = n >> 1;
        const float th0 = -6.28318530717958647692f / (float)n;
        for (int j = tid; j < NFFT / 2; j += 128) {
            int p = j >> stage;           // j / s
            int q = j & (s - 1);          // j % s
            float ang = th0 * (float)p;
            float wr = __cosf(ang), wi = __sinf(ang);
            float2 a = X[q + s * p];
            float2 c = X[q + s * (p + m)];
            Y[q + s * (2 * p)]     = make_float2(a.x + c.x, a.y + c.y);
            float dx = a.x - c.x, dy = a.y - c.y;
            Y[q + s * (2 * p + 1)] = make_float2(dx * wr - dy * wi,
                                                 dx * wi + dy * wr);
        }
        __syncthreads();
        float2* tp = X; X = Y; Y = tp;
        n >>= 1; s <<= 1;
    }

    // power spectrum, bins 0..512; pad 513..519 with zero
    float* out = spec + (size_t)fr * FPAD;
    for (int k = tid; k < FPAD; k += 128) {
        float pw = 0.0f;
        if (k <= NFFT / 2) { float2 v = X[k]; pw = v.x * v.x + v.y * v.y; }
        out[k] = pw;
    }
}

// ---------------------------------------------------------------------------
// Kernel 2: generic f32 WMMA GEMM.  C[m][n<40] = sum_k A[m][k] * B[k][n]
// A: M x K (row stride lda), B: K x NPAD (zero padded), C stride NFILT.
// The block's 16-column B panel (K x 16, row stride NPAD) is staged into LDS
// with one Tensor-Data-Mover transfer (tensor_load_to_lds, TENSORcnt), then
// 4 waves each compute one 16x16 tile via V_WMMA_F32_16X16X4_F32.
// Lane mapping per CDNA5 ISA 7.12.2:
//   A 16x4: lanes 0-15 {K0,K1}, lanes 16-31 {K2,K3}, M = lane&15
//   B 4x16: lanes 0-15 {K0,K1}, lanes 16-31 {K2,K3}, N = lane&15
//   C/D:    VGPR i -> M = i + 8*(lane>>4), N = lane&15
// ---------------------------------------------------------------------------
__global__ void __launch_bounds__(128)
wmma_gemm_f32_kernel(const float* __restrict__ A, int lda, int K,
                     const float* __restrict__ Bm,
                     float* __restrict__ C, int mtiles)
{
    __shared__ float Bs[KMAX * 16];          // staged B panel: Bs[k*16 + n]

    const int wave = threadIdx.x >> 5;
    const int lane = threadIdx.x & 31;
    const int n0   = blockIdx.y * 16;

    // ---- TDM: DMA the K x 16 B panel (row stride NPAD floats) into LDS ----
    if (wave == 0) {                          // wave-uniform; TDM ignores EXEC
        unsigned long long ga = (unsigned long long)(uintptr_t)(Bm + n0);
        unsigned lds = (unsigned)(uintptr_t)&Bs[0];     // low 32b = LDS offset
        // D# group0: count=1 | lds_addr | global_addr[56:0] | type=2
        u32x4 g0 = { 1u, lds, (unsigned)ga,
                     ((unsigned)(ga >> 32) & 0x01FFFFFFu) | 0x80000000u };
        // D# group1: data_size=4B(code2); tensor_dim0=NPAD; tensor_dim1=K;
        //            tile_dim0=16; tile_dim1=K; tensor_dim0_stride=NPAD
        i32x8 g1 = { (int)0x00020000u,
                     (int)((unsigned)NPAD << 16),
                     (int)((unsigned)K << 16),
                     (int)(16u << 16),
                     K,
                     NPAD,
                     0, 0 };
        i32x4 gz = { 0, 0, 0, 0 };            // groups 2/3 unused (2D tile)
#if defined(__clang_major__) && (__clang_major__ >= 23)
        i32x8 gz8 = { 0, 0, 0, 0, 0, 0, 0, 0 };
        __builtin_amdgcn_tensor_load_to_lds(g0, g1, gz, gz, gz8, 0);
#else
        __builtin_amdgcn_tensor_load_to_lds(g0, g1, gz, gz, 0);
#endif
        __builtin_amdgcn_s_wait_tensorcnt(0);
    }
    __syncthreads();                          // publish LDS to all 4 waves

    const int mt = blockIdx.x * 4 + wave;
    if (mt < mtiles) {                        // wave-uniform: EXEC stays all-1s
        const int l15  = lane & 15;
        const int half = lane >> 4;           // 0: K{0,1}, 1: K{2,3}
        const float* Ap = A + (size_t)(mt * 16 + l15) * lda + 2 * half;
        const float* Bp = &Bs[(size_t)(2 * half) * 16 + l15];

        v8f acc = {};
        for (int k = 0; k < K; k += 4) {
            if ((k & 15) == 0) __builtin_prefetch(Ap + k + 64, 0, 1);
            v2f a, bv;
            a.x  = Ap[k];
            a.y  = Ap[k + 1];
            bv.x = Bp[k * 16];
            bv.y = Bp[k * 16 + 16];
            acc = __builtin_amdgcn_wmma_f32_16x16x4_f32(
                      false, a, false, bv, (short)0, acc, false, false);
        }

        const int ncol = n0 + l15;
        if (ncol < NFILT) {
            const int mbase = mt * 16 + half * 8;
            #pragma unroll
            for (int i = 0; i < 8; ++i)
                C[(size_t)(mbase + i) * NFILT + ncol] = acc[i];
        }
    }
}

// ---------------------------------------------------------------------------
// B-matrix preparation (transpose + zero pad to NPAD columns)
// ---------------------------------------------------------------------------
__global__ void build_bt_fbank_kernel(const float* __restrict__ fb,
                                      float* __restrict__ Bt)
{
    int idx = blockIdx.x * blockDim.x + threadIdx.x;
    if (idx >= FPAD * NPAD) return;
    int f = idx / NPAD, c = idx % NPAD;
    Bt[idx] = (f < FBINS && c < NFILT) ? fb[(size_t)c * FBINS + f] : 0.0f;
}

__global__ void build_bt_dct_kernel(const float* __restrict__ dct,
                                    float* __restrict__ Bt)
{
    int idx = blockIdx.x * blockDim.x + threadIdx.x;
    if (idx >= NFILT * NPAD) return;
    int c = idx / NPAD, n = idx % NPAD;
    Bt[idx] = (n < NFILT) ? dct[(size_t)n * NFILT + c] : 0.0f;   // B[c][n]=dct[n][c]
}

// ---------------------------------------------------------------------------
// Kernel 3: medium-time power Qt = 5-wide zero-padded time mean of P
// ---------------------------------------------------------------------------
__global__ void qt_kernel(const float* __restrict__ P, float* __restrict__ Qt)
{
    int idx = blockIdx.x * blockDim.x + threadIdx.x;
    if (idx >= MROWS * NFILT) return;
    int c  = idx % NFILT;
    int bt = idx / NFILT;
    int t  = bt % NFRAMES;
    int b  = bt / NFRAMES;
    float s = 0.0f;
    #pragma unroll
    for (int dt = -2; dt <= 2; ++dt) {
        int tt = t + dt;
        if (tt >= 0 && tt < NFRAMES)
            s += P[((size_t)b * NFRAMES + tt) * NFILT + c];
    }
    Qt[idx] = s * 0.2f;
}

// ---------------------------------------------------------------------------
// Kernel 4: fused sequential scans over frames, one thread per (b, channel):
//   Qle = alp(P); Q0 = relu(P - Qle); Qf = alp(Q0); Qtm = temporal_mask(Q0);
//   R = (P >= CC*Qle) ? Qtm : Qf
// ---------------------------------------------------------------------------
__global__ void scan_kernel(const float* __restrict__ P, float* __restrict__ R)
{
    int gid = blockIdx.x * blockDim.x + threadIdx.x;
    if (gid >= BATCH * NFILT) return;
    int b = gid / NFILT;
    int c = gid % NFILT;

    float prevA = 0.0f, prevF = 0.0f, qp = 0.0f;
    for (int t = 0; t < NFRAMES; ++t) {
        size_t o = ((size_t)b * NFRAMES + t) * NFILT + c;
        float p = P[o];

        float qle = (p > prevA) ? (LM_A * prevA + (1.0f - LM_A) * p)
                                : (LM_B * prevA + (1.0f - LM_B) * p);
        prevA = qle;

        float q0 = fmaxf(p - qle, 0.0f);

        float qf = (q0 > prevF) ? (LM_A * prevF + (1.0f - LM_A) * q0)
                                : (LM_B * prevF + (1.0f - LM_B) * q0);
        prevF = qf;

        float qtm;
        if (t == 0) { qtm = q0; qp = q0; }
        else {
            qtm = (q0 >= LM_T * qp) ? q0 : MU_T * qp;
            qp  = fmaxf(LM_T * qp, q0);
        }

        R[o] = (p >= CC * qle) ? qtm : qf;
    }
}

// ---------------------------------------------------------------------------
// Kernel 5: weight smoothing over channels (9-wide, edge-count normalized)
// then T = P * St
// ---------------------------------------------------------------------------
__global__ void smooth_kernel(const float* __restrict__ P,
                              const float* __restrict__ R,
                              const float* __restrict__ Qt,
                              float* __restrict__ Tb)
{
    int idx = blockIdx.x * blockDim.x + threadIdx.x;
    if (idx >= MROWS * NFILT) return;
    int c  = idx % NFILT;
    int bt = idx / NFILT;
    int lo = (c - 4 < 0) ? 0 : c - 4;
    int hi = (c + 4 > NFILT - 1) ? NFILT - 1 : c + 4;
    float s = 0.0f;
    for (int l2 = lo; l2 <= hi; ++l2) {
        size_t o = (size_t)bt * NFILT + l2;
        s += R[o] / Qt[o];
    }
    float st = s / (float)(hi - lo + 1);
    Tb[idx] = P[idx] * st;
}

// ---------------------------------------------------------------------------
// Kernel 6: Tf = channel mean of T
// ---------------------------------------------------------------------------
__global__ void tf_kernel(const float* __restrict__ Tb, float* __restrict__ Tf)
{
    int bt = blockIdx.x * blockDim.x + threadIdx.x;
    if (bt >= MROWS) return;
    float s = 0.0f;
    const float* row = Tb + (size_t)bt * NFILT;
    #pragma unroll
    for (int c = 0; c < NFILT; ++c) s += row[c];
    Tf[bt] = s * (1.0f / NFILT);
}

// ---------------------------------------------------------------------------
// Kernel 7: running-mean scan:  mu[0]=1e-4; mu[t]=0.999 mu[t-1]+0.001 Tf[t]
// ---------------------------------------------------------------------------
__global__ void mu_kernel(const float* __restrict__ Tf, float* __restrict__ Mu)
{
    int b = threadIdx.x;
    if (b >= BATCH) return;
    float mu = 1e-4f;
    Mu[(size_t)b * NFRAMES] = mu;
    for (int t = 1; t < NFRAMES; ++t) {
        mu = LM_MU * mu + (1.0f - LM_MU) * Tf[(size_t)b * NFRAMES + t];
        Mu[(size_t)b * NFRAMES + t] = mu;
    }
}

// ---------------------------------------------------------------------------
// Kernel 8: V = (T / mu)^(1/15)
// ---------------------------------------------------------------------------
__global__ void v_kernel(const float* __restrict__ Tb,
                         const float* __restrict__ Mu,
                         float* __restrict__ Vb)
{
    int idx = blockIdx.x * blockDim.x + threadIdx.x;
    if (idx >= MROWS * NFILT) return;
    int bt = idx / NFILT;
    Vb[idx] = powf(Tb[idx] / Mu[bt], 1.0f / 15.0f);
}

// ---------------------------------------------------------------------------
// kernel_launch: orchestrate the pipeline on `stream`.
// ws layout (floats):
//   [0, MROWS*FPAD)            spec   (reused for Qt/R/T/V/Tf/Mu afterwards)
//   [+MROWS*FPAD, +MROWS*40)   P
//   [.., +FPAD*48)             Bt1 (fbank^T padded)
//   [.., +40*48)               Bt2 (dct^T padded)
// total ~72 MB
// ---------------------------------------------------------------------------
extern "C" void kernel_launch(void* const* d_in, const int* in_sizes, int n_in,
                              void* d_out, int out_size, void* d_ws, size_t ws_size,
                              hipStream_t stream)
{
    (void)in_sizes; (void)n_in; (void)out_size; (void)ws_size;

    const float* x   = (const float*)d_in[0];
    const float* win = (const float*)d_in[1];
    const float* fb  = (const float*)d_in[2];
    const float* dct = (const float*)d_in[3];
    float* out = (float*)d_out;
    float* ws  = (float*)d_ws;

    float* spec = ws;
    float* Pm   = ws + (size_t)MROWS * FPAD;
    float* Bt1  = Pm + (size_t)MROWS * NFILT;
    float* Bt2  = Bt1 + (size_t)FPAD * NPAD;

    // reuse of spec region once P exists (5.2 MB slabs inside 66.6 MB)
    float* Qt = ws;
    float* R  = ws + 1 * (size_t)MROWS * NFILT;
    float* Tb = ws + 2 * (size_t)MROWS * NFILT;
    float* Vb = ws + 3 * (size_t)MROWS * NFILT;
    float* Tf = ws + 4 * (size_t)MROWS * NFILT;
    float* Mu = Tf + MROWS;

    stft_power_kernel<<<MROWS, 128, 0, stream>>>(x, win, spec);

    build_bt_fbank_kernel<<<(FPAD * NPAD + 255) / 256, 256, 0, stream>>>(fb, Bt1);
    build_bt_dct_kernel<<<(NFILT * NPAD + 255) / 256, 256, 0, stream>>>(dct, Bt2);

    dim3 gTiles((MTILES + 3) / 4, NPAD / 16);
    wmma_gemm_f32_kernel<<<gTiles, 128, 0, stream>>>(spec, FPAD, FPAD, Bt1, Pm, MTILES);

    const int nel = MROWS * NFILT;
    qt_kernel<<<(nel + 255) / 256, 256, 0, stream>>>(Pm, Qt);
    scan_kernel<<<(BATCH * NFILT + 127) / 128, 128, 0, stream>>>(Pm, R);
    smooth_kernel<<<(nel + 255) / 256, 256, 0, stream>>>(Pm, R, Qt, Tb);
    tf_kernel<<<(MROWS + 255) / 256, 256, 0, stream>>>(Tb, Tf);
    mu_kernel<<<1, 32, 0, stream>>>(Tf, Mu);
    v_kernel<<<(nel + 255) / 256, 256, 0, stream>>>(Tb, Mu, Vb);

    wmma_gemm_f32_kernel<<<gTiles, 128, 0, stream>>>(Vb, NFILT, NFILT, Bt2, out, MTILES);
}